// TMetaLSTM_52871047413865
// MI455X (gfx1250) — compile-verified
//
#include <hip/hip_runtime.h>
#include <hip/hip_bf16.h>
#include <math.h>

// ---------------- problem constants ----------------
#define BB    64
#define TT    12
#define NN    325
#define HH    64
#define EE    32
#define LHH   128
#define OUTC  12
#define XFEAT 97      // 1 + 64 + 32
#define METAOFF 65
#define NPAD  336     // 21 * 16
#define NTILES 21
#define BLOCK 672     // 21 waves of 32

typedef float v2f __attribute__((ext_vector_type(2)));
typedef float v8f __attribute__((ext_vector_type(8)));

__device__ __forceinline__ float sigmoidf(float x) {
    return 1.0f / (1.0f + expf(-x));
}

// ---------------- stage 1: meta = mean_t x[:,t,0,65:] ----------------
__global__ void k_meta(const float* __restrict__ x, float* __restrict__ meta) {
    int b = blockIdx.x;
    int e = threadIdx.x;                 // 32 threads
    float s = 0.f;
    for (int t = 0; t < TT; ++t)
        s += x[((size_t)(b * TT + t) * NN + 0) * XFEAT + METAOFF + e];
    meta[b * EE + e] = s * (1.0f / TT);
}

// ---------------- stage 2a: hidden = relu(meta @ W1 + b1) for 3 MLPs x 4 gates
// hid layout: (3,4,B,128)
__global__ void k_hidden(const float* __restrict__ meta,
                         const float* __restrict__ W1x, const float* __restrict__ b1x,
                         const float* __restrict__ W1h, const float* __restrict__ b1h,
                         const float* __restrict__ W1b, const float* __restrict__ b1b,
                         float* __restrict__ hid) {
    int m  = blockIdx.y;                 // which MLP: 0=lwx 1=lwh 2=lb
    int gb = blockIdx.x;                 // 4*B
    int g  = gb >> 6, b = gb & 63;
    int j  = threadIdx.x;                // 128
    const float* W1 = (m == 0) ? W1x : ((m == 1) ? W1h : W1b);
    const float* b1 = (m == 0) ? b1x : ((m == 1) ? b1h : b1b);
    float acc = b1[g * LHH + j];
    for (int e = 0; e < EE; ++e)
        acc += meta[b * EE + e] * W1[(g * EE + e) * LHH + j];
    hid[((m * 4 + g) * BB + b) * LHH + j] = fmaxf(acc, 0.f);
}

// ---------------- stage 2b: Wx (4,B,64) and bvec (4,B,64) ----------------
__global__ void k_small_out(const float* __restrict__ hid,
                            const float* __restrict__ W2x, const float* __restrict__ b2x,
                            const float* __restrict__ W2b, const float* __restrict__ b2b,
                            float* __restrict__ Wx, float* __restrict__ bvec) {
    int which = blockIdx.y;              // 0 -> Wx (hid mlp 0), 1 -> bvec (hid mlp 2)
    int gb = blockIdx.x;
    int g = gb >> 6, b = gb & 63;
    int k = threadIdx.x;                 // 64
    const float* hv = &hid[(((which ? 2 : 0) * 4 + g) * BB + b) * LHH];
    const float* W2 = which ? W2b : W2x;
    const float* b2 = which ? b2b : b2x;
    float acc = b2[g * HH + k];
    for (int j = 0; j < LHH; ++j)
        acc += hv[j] * W2[(g * LHH + j) * HH + k];
    float* dst = which ? bvec : Wx;
    dst[(g * BB + b) * HH + k] = acc;
}

// ---------------- stage 2c: WhT (4,B,64,64), stored TRANSPOSED [k][h] -------
__global__ void k_whT(const float* __restrict__ hid,
                      const float* __restrict__ W2h, const float* __restrict__ b2h,
                      float* __restrict__ WhT) {
    int gb = blockIdx.x;
    int g = gb >> 6, b = gb & 63;
    const float* hv = &hid[((1 * 4 + g) * BB + b) * LHH];
    for (int o = threadIdx.x; o < HH * HH; o += blockDim.x) {
        float acc = b2h[g * HH * HH + o];
        for (int j = 0; j < LHH; ++j)
            acc += hv[j] * W2h[(size_t)(g * LHH + j) * (HH * HH) + o];
        int hi = o >> 6, k = o & 63;     // reference: flat o = h*H + k
        WhT[(size_t)((g * BB + b) * HH + k) * HH + hi] = acc;  // transposed
    }
}

// ---------------- stage 3: LSTM recurrence + head, one block per batch -------
// LDS: WhT 4*64*64 | h 336*64 | xt 336 | Wx 4*64 | b 4*64
#define SM_WHT   0
#define SM_H     (4 * HH * HH)                 // 16384
#define SM_XT    (SM_H + NPAD * HH)            // 16384 + 21504 = 37888
#define SM_WX    (SM_XT + NPAD)                // 38224
#define SM_B     (SM_WX + 4 * HH)              // 38480
#define SM_TOTAL (SM_B + 4 * HH)               // 38736 floats = 154944 bytes

__global__ void __launch_bounds__(BLOCK)
k_lstm(const float* __restrict__ x,
       const float* __restrict__ Wxg,   // (4,B,64)
       const float* __restrict__ WhTg,  // (4,B,64,64) transposed [k][h]
       const float* __restrict__ bg,    // (4,B,64)
       const float* __restrict__ fc1W, const float* __restrict__ fc1b,
       const float* __restrict__ fc2W, const float* __restrict__ fc2b,
       float* __restrict__ out) {
    extern __shared__ float smem[];
    float* sWhT = smem + SM_WHT;
    float* sH   = smem + SM_H;
    float* sXt  = smem + SM_XT;
    float* sWx  = smem + SM_WX;
    float* sB   = smem + SM_B;

    const int b    = blockIdx.x;
    const int tid  = threadIdx.x;
    const int lane = tid & 31;
    const int wave = tid >> 5;           // 0..20 == row tile
    const int half = lane >> 4;          // 0: lanes 0-15, 1: lanes 16-31
    const int lm   = lane & 15;
    const int rowbase = wave * 16;

    // ---- stage per-batch weights into LDS ----
    for (int i = tid; i < 4 * HH * HH; i += BLOCK) {
        int g = i >> 12, o = i & 4095;
        sWhT[i] = WhTg[(size_t)(g * BB + b) * (HH * HH) + o];
    }
    for (int i = tid; i < 4 * HH; i += BLOCK) {
        int g = i >> 6, k = i & 63;
        sWx[i] = Wxg[(g * BB + b) * HH + k];
        sB[i]  = bg[(g * BB + b) * HH + k];
    }
    for (int i = tid; i < NPAD * HH; i += BLOCK) sH[i] = 0.f;   // h0 = 0
    __syncthreads();

    // per-thread recurrent state, C-layout fragments: (v, lane) -> row v+8*half, col 16*ct+lm
    float creg[4][8];
    float hnew[4][8];
#pragma unroll
    for (int ct = 0; ct < 4; ++ct)
#pragma unroll
        for (int v = 0; v < 8; ++v) creg[ct][v] = 0.f;

#pragma unroll 1
    for (int t = 0; t < TT; ++t) {
        // stage xt for this wave's 16 rows (x[b,t,n,0])
        if (lm == lane && lm < 16) { } // no-op keep structure simple
        if (lane < 16) {
            int n = rowbase + lane;
            float v = 0.f;
            if (n < NN) v = x[((size_t)(b * TT + t) * NN + n) * XFEAT];
            sXt[n] = v;
        }
        __syncthreads();   // also orders previous step's sH writeback vs. reads

        float xtv[8];
#pragma unroll
        for (int v = 0; v < 8; ++v) xtv[v] = sXt[rowbase + 8 * half + v];

#pragma unroll 1
        for (int ct = 0; ct < 4; ++ct) {
            const int kcol = 16 * ct + lm;
            // init z = xt * Wx + b  (outer product since D==1), C-fragment layout
            v8f acc[4];
#pragma unroll
            for (int g = 0; g < 4; ++g) {
                float wxk = sWx[g * HH + kcol];
                float bk  = sB[g * HH + kcol];
#pragma unroll
                for (int v = 0; v < 8; ++v) acc[g][v] = xtv[v] * wxk + bk;
            }
            // z += h @ Wh[g]  via V_WMMA_F32_16X16X4_F32, K = 64 -> 16 steps
#pragma unroll
            for (int kk = 0; kk < 16; ++kk) {
                // A frag: rows = this tile's h rows, K pair chosen by lane half
                v2f a = *(const v2f*)&sH[(rowbase + lm) * HH + 4 * kk + 2 * half];
#pragma unroll
                for (int g = 0; g < 4; ++g) {
                    v2f bf = *(const v2f*)&sWhT[g * (HH * HH) + kcol * HH + 4 * kk + 2 * half];
                    acc[g] = __builtin_amdgcn_wmma_f32_16x16x4_f32(
                        false, a, false, bf, (short)0, acc[g], false, false);
                }
            }
            // gate nonlinearities + cell update (8 elems per thread)
#pragma unroll
            for (int v = 0; v < 8; ++v) {
                float gg = tanhf(acc[0][v]);
                float ii = sigmoidf(acc[1][v]);
                float ff = sigmoidf(acc[2][v]);
                float oo = sigmoidf(acc[3][v]);
                float c  = gg * ii + creg[ct][v] * ff;
                creg[ct][v] = c;
                hnew[ct][v] = tanhf(c) * oo;
            }
        }
        // write back new h AFTER all col tiles consumed old h
#pragma unroll
        for (int ct = 0; ct < 4; ++ct)
#pragma unroll
            for (int v = 0; v < 8; ++v)
                sH[(rowbase + v + 8 * half) * HH + 16 * ct + lm] = hnew[ct][v];
        __syncthreads();
    }

    // ---- head: relu(h) -> fc1(64->32) relu -> fc2(32->12) ----
    if (tid < NN) {
        const int n = tid;
        float hid2[HH / 2];
#pragma unroll
        for (int m = 0; m < HH / 2; ++m) hid2[m] = fc1b[m];
        for (int j = 0; j < HH; ++j) {
            float rv = fmaxf(sH[n * HH + j], 0.f);
#pragma unroll
            for (int m = 0; m < HH / 2; ++m)
                hid2[m] += rv * fc1W[j * (HH / 2) + m];
        }
#pragma unroll
        for (int m = 0; m < HH / 2; ++m) hid2[m] = fmaxf(hid2[m], 0.f);
#pragma unroll
        for (int oc = 0; oc < OUTC; ++oc) {
            float a = fc2b[oc];
#pragma unroll
            for (int m = 0; m < HH / 2; ++m) a += hid2[m] * fc2W[m * OUTC + oc];
            out[((size_t)b * OUTC + oc) * NN + n] = a;   // (B,OUT,N,1)
        }
    }
}

// ---------------- workspace layout (floats) ----------------
#define WS_META  0
#define WS_HID   (WS_META + BB * EE)                       // 2048
#define WS_WX    (WS_HID + 3 * 4 * BB * LHH)               // 2048 + 98304
#define WS_BVEC  (WS_WX + 4 * BB * HH)                     // + 16384
#define WS_WHT   (WS_BVEC + 4 * BB * HH)                   // + 16384
#define WS_END   (WS_WHT + 4 * BB * HH * HH)               // + 1048576

extern "C" void kernel_launch(void* const* d_in, const int* in_sizes, int n_in,
                              void* d_out, int out_size, void* d_ws, size_t ws_size,
                              hipStream_t stream) {
    const float* x      = (const float*)d_in[0];
    const float* lwx_W1 = (const float*)d_in[1];
    const float* lwx_b1 = (const float*)d_in[2];
    const float* lwx_W2 = (const float*)d_in[3];
    const float* lwx_b2 = (const float*)d_in[4];
    const float* lwh_W1 = (const float*)d_in[5];
    const float* lwh_b1 = (const float*)d_in[6];
    const float* lwh_W2 = (const float*)d_in[7];
    const float* lwh_b2 = (const float*)d_in[8];
    const float* lb_W1  = (const float*)d_in[9];
    const float* lb_b1  = (const float*)d_in[10];
    const float* lb_W2  = (const float*)d_in[11];
    const float* lb_b2  = (const float*)d_in[12];
    const float* fc1_W  = (const float*)d_in[13];
    const float* fc1_b  = (const float*)d_in[14];
    const float* fc2_W  = (const float*)d_in[15];
    const float* fc2_b  = (const float*)d_in[16];
    float* out = (float*)d_out;
    float* ws  = (float*)d_ws;

    float* meta = ws + WS_META;
    float* hid  = ws + WS_HID;
    float* Wx   = ws + WS_WX;
    float* bvec = ws + WS_BVEC;
    float* WhT  = ws + WS_WHT;

    k_meta<<<BB, EE, 0, stream>>>(x, meta);
    k_hidden<<<dim3(4 * BB, 3), LHH, 0, stream>>>(meta, lwx_W1, lwx_b1,
                                                  lwh_W1, lwh_b1, lb_W1, lb_b1, hid);
    k_small_out<<<dim3(4 * BB, 2), HH, 0, stream>>>(hid, lwx_W2, lwx_b2,
                                                    lb_W2, lb_b2, Wx, bvec);
    k_whT<<<4 * BB, 256, 0, stream>>>(hid, lwh_W2, lwh_b2, WhT);
    k_lstm<<<BB, BLOCK, SM_TOTAL * sizeof(float), stream>>>(
        x, Wx, WhT, bvec, fc1_W, fc1_b, fc2_W, fc2_b, out);
}